// PhACE_79276506349981
// MI455X (gfx1250) — compile-verified
//
#include <hip/hip_runtime.h>
#include <math.h>

typedef __attribute__((ext_vector_type(2))) float v2f;
typedef __attribute__((ext_vector_type(8))) float v8f;

#define EDGES_PER_BLOCK 8

// ---------------------------------------------------------------------------
// Per-edge geometry: radial basis (gaussian * cosine cutoff) + real SH l<=3
// ---------------------------------------------------------------------------
__device__ __forceinline__ void edge_geometry(const float* __restrict__ pos,
                                              int s, int rc,
                                              float* R, float* Y) {
  float dx = pos[3 * rc + 0] - pos[3 * s + 0];
  float dy = pos[3 * rc + 1] - pos[3 * s + 1];
  float dz = pos[3 * rc + 2] - pos[3 * s + 2];
  float r = sqrtf(dx * dx + dy * dy + dz * dz + 1e-12f);
  float ir = 1.0f / r;
  float x = dx * ir, y = dy * ir, z = dz * ir;
  float fc = (r < 5.0f) ? 0.5f * (cosf(3.14159265358979323f * r * 0.2f) + 1.0f)
                        : 0.0f;
#pragma unroll
  for (int j = 0; j < 8; j++) {
    float mu = (5.0f / 7.0f) * (float)j;   // linspace(0, 5, 8)
    float t = r - mu;
    R[j] = expf(-t * t) * fc;
  }
  const float s3  = 1.7320508075688772f;   // sqrt(3)
  const float c33 = 0.7905694150420949f;   // sqrt(10)/4
  const float c31 = 0.6123724356957945f;   // sqrt(6)/4
  const float s15 = 3.8729833462074170f;   // sqrt(15)
  Y[0]  = 1.0f;
  Y[1]  = x; Y[2] = y; Y[3] = z;
  Y[4]  = s3 * x * y;
  Y[5]  = s3 * y * z;
  Y[6]  = 0.5f * (3.0f * z * z - 1.0f);
  Y[7]  = s3 * x * z;
  Y[8]  = 0.5f * s3 * (x * x - y * y);
  Y[9]  = c33 * y * (3.0f * x * x - y * y);
  Y[10] = s15 * x * y * z;
  Y[11] = c31 * y * (5.0f * z * z - 1.0f);
  Y[12] = 0.5f * z * (5.0f * z * z - 3.0f);
  Y[13] = c31 * x * (5.0f * z * z - 1.0f);
  Y[14] = 0.5f * s15 * (x * x - y * y) * z;
  Y[15] = c33 * x * (x * x - 3.0f * y * y);
}

// ---------------------------------------------------------------------------
// Pass 1: A[l][recv][m][k] += MP * (R@Wr[l])[k] * emb_tile[send][k] * Y[l][m]
// One wave32 per edge; lanes sweep k (coalesced LDS reads + coalesced atomics,
// A (38 MB) is L2-resident on MI455X's 192 MB L2).
// ---------------------------------------------------------------------------
__global__ void edge_pass1(const float* __restrict__ pos,
                           const float* __restrict__ emb,
                           const int* __restrict__ species,
                           const int* __restrict__ senders,
                           const int* __restrict__ receivers,
                           const float* __restrict__ Wr0,
                           const float* __restrict__ Wr1,
                           const float* __restrict__ Wr2,
                           const float* __restrict__ Wr3,
                           float* __restrict__ Abase,
                           int n_atoms, int n_edges) {
  __shared__ float sW[2560];  // 8 x (128+96+64+32)
  __shared__ float sE[64];    // 4 species x 16 channels
  for (int i = threadIdx.x; i < 1024; i += 256) sW[i]        = Wr0[i];
  for (int i = threadIdx.x; i < 768;  i += 256) sW[1024 + i] = Wr1[i];
  for (int i = threadIdx.x; i < 512;  i += 256) sW[1792 + i] = Wr2[i];
  for (int i = threadIdx.x; i < 256;  i += 256) sW[2304 + i] = Wr3[i];
  if (threadIdx.x < 64) sE[threadIdx.x] = emb[threadIdx.x];
  __syncthreads();

  int lane = threadIdx.x & 31;
  int e = blockIdx.x * EDGES_PER_BLOCK + (threadIdx.x >> 5);
  if (e >= n_edges) return;
  int s = senders[e], rc = receivers[e];

  float R[8], Y[16];
  edge_geometry(pos, s, rc, R, Y);
  int spb = species[s] * 16;

  const int Ks[4] = {128, 96, 64, 32};
  const int Nm[4] = {8, 6, 4, 2};
  const int Wo[4] = {0, 1024, 1792, 2304};
  const int Yo[4] = {0, 1, 4, 9};

  size_t Ao = 0;
#pragma unroll
  for (int l = 0; l < 4; l++) {
    const int Kl = Ks[l];
    const int rows = 2 * l + 1;
    float* Al = Abase + Ao + (size_t)rc * (size_t)(rows * Kl);
    const float* w = &sW[Wo[l]];
    for (int k0 = 0; k0 < Kl; k0 += 32) {   // Kl is a multiple of 32
      int k = k0 + lane;
      float rl = 0.0f;
#pragma unroll
      for (int j = 0; j < 8; j++) rl += R[j] * w[j * Kl + k];
      float base = 0.1f * rl * sE[spb + k / Nm[l]];  // MP_SCALING = 0.1
#pragma unroll
      for (int m = 0; m < rows; m++)
        atomicAdd(&Al[m * Kl + k], base * Y[Yo[l] + m]);
    }
    Ao += (size_t)n_atoms * (size_t)(rows * Kl);
  }
}

// ---------------------------------------------------------------------------
// Pass 2: B[l][recv][m][k] += MP * A[l][send][m][k] * emb_tile[send][k]
//                                * (R@Vr[l])[k]           (no Y factor)
// ---------------------------------------------------------------------------
__global__ void edge_pass2(const float* __restrict__ pos,
                           const float* __restrict__ emb,
                           const int* __restrict__ species,
                           const int* __restrict__ senders,
                           const int* __restrict__ receivers,
                           const float* __restrict__ Vr0,
                           const float* __restrict__ Vr1,
                           const float* __restrict__ Vr2,
                           const float* __restrict__ Vr3,
                           const float* __restrict__ Abase,
                           float* __restrict__ Bbase,
                           int n_atoms, int n_edges) {
  __shared__ float sW[2560];
  __shared__ float sE[64];
  for (int i = threadIdx.x; i < 1024; i += 256) sW[i]        = Vr0[i];
  for (int i = threadIdx.x; i < 768;  i += 256) sW[1024 + i] = Vr1[i];
  for (int i = threadIdx.x; i < 512;  i += 256) sW[1792 + i] = Vr2[i];
  for (int i = threadIdx.x; i < 256;  i += 256) sW[2304 + i] = Vr3[i];
  if (threadIdx.x < 64) sE[threadIdx.x] = emb[threadIdx.x];
  __syncthreads();

  int lane = threadIdx.x & 31;
  int e = blockIdx.x * EDGES_PER_BLOCK + (threadIdx.x >> 5);
  if (e >= n_edges) return;
  int s = senders[e], rc = receivers[e];

  float R[8], Yunused[16];
  edge_geometry(pos, s, rc, R, Yunused);
  int spb = species[s] * 16;

  const int Ks[4] = {128, 96, 64, 32};
  const int Nm[4] = {8, 6, 4, 2};
  const int Wo[4] = {0, 1024, 1792, 2304};

  size_t Ao = 0;
#pragma unroll
  for (int l = 0; l < 4; l++) {
    const int Kl = Ks[l];
    const int rows = 2 * l + 1;
    const float* Al = Abase + Ao + (size_t)s  * (size_t)(rows * Kl);
    float*       Bl = Bbase + Ao + (size_t)rc * (size_t)(rows * Kl);
    const float* w = &sW[Wo[l]];
    for (int k0 = 0; k0 < Kl; k0 += 32) {
      int k = k0 + lane;
      float vl = 0.0f;
#pragma unroll
      for (int j = 0; j < 8; j++) vl += R[j] * w[j * Kl + k];
      float hv = 0.1f * vl * sE[spb + k / Nm[l]];
#pragma unroll
      for (int m = 0; m < rows; m++)
        atomicAdd(&Bl[m * Kl + k], Al[m * Kl + k] * hv);
    }
    Ao += (size_t)n_atoms * (size_t)(rows * Kl);
  }
}

// ---------------------------------------------------------------------------
// inv[i][j] = sum_m B_l[i][m][k]^2 / sqrt(2l+1)  for j over concat(l=1..3)
// ---------------------------------------------------------------------------
__global__ void inv_kernel(const float* __restrict__ Bbase,
                           float* __restrict__ inv, int n_atoms) {
  int idx = blockIdx.x * blockDim.x + threadIdx.x;
  if (idx >= n_atoms * 192) return;
  int i = idx / 192;
  int j = idx - i * 192;
  int l, kl;
  if (j < 96)       { l = 1; kl = j; }
  else if (j < 160) { l = 2; kl = j - 96; }
  else              { l = 3; kl = j - 160; }
  const int Ks[4] = {128, 96, 64, 32};
  size_t off = (size_t)n_atoms * 128;                    // skip l=0
  if (l >= 2) off += (size_t)n_atoms * 288;
  if (l == 3) off += (size_t)n_atoms * 320;
  int rows = 2 * l + 1, Kl = Ks[l];
  const float* Bl = Bbase + off + (size_t)i * (size_t)(rows * Kl);
  float acc = 0.0f;
  for (int m = 0; m < rows; m++) {
    float v = Bl[m * Kl + kl];
    acc += v * v;
  }
  const float isq[4] = {1.0f, 0.57735026918962576f, 0.44721359549995794f,
                        0.37796447300922720f};
  inv[(size_t)i * 192 + j] = acc * isq[l];
}

// ---------------------------------------------------------------------------
// Concatenate N1[96,128], N2[64,128], N3[32,128] -> Ncat[192,128]
// ---------------------------------------------------------------------------
__global__ void pack_ncat(const float* __restrict__ N1,
                          const float* __restrict__ N2,
                          const float* __restrict__ N3,
                          float* __restrict__ ncat) {
  int idx = blockIdx.x * blockDim.x + threadIdx.x;
  if (idx >= 192 * 128) return;
  int rrow = idx >> 7;
  int col = idx & 127;
  float v;
  if (rrow < 96)       v = N1[rrow * 128 + col];
  else if (rrow < 160) v = N2[(rrow - 96) * 128 + col];
  else                 v = N3[(rrow - 160) * 128 + col];
  ncat[idx] = v;
}

// ---------------------------------------------------------------------------
// fp32 GEMM via V_WMMA_F32_16X16X4_F32: D = A[MxK] * B[KxNcols] (+Cinit),
// optional fused SiLU. One wave per 16x16 output tile; 8 waves/block cover a
// 16-row strip across Ncols=128. A-frag: lanes 0-15 carry K={kk,kk+1},
// lanes 16-31 K={kk+2,kk+3}; B/C/D row-striped across lanes per ISA layout.
// ---------------------------------------------------------------------------
__global__ void wmma_gemm_f32(const float* __restrict__ Amat, int lda, int kdim,
                              const float* __restrict__ Bmat, int ldb,
                              const float* __restrict__ Cinit,
                              float* __restrict__ Dmat,
                              int M, int ncols, int apply_silu) {
  int lane = threadIdx.x & 31;
  int g = lane >> 4;       // half-wave select (K pair / row half)
  int lr = lane & 15;
  int tileM = blockIdx.x * 16;
  int tileN = (threadIdx.x >> 5) * 16;
  if (tileN >= ncols) return;          // wave-uniform
  int col = tileN + lr;

  v8f c;
#pragma unroll
  for (int rIt = 0; rIt < 8; rIt++) {
    int row = tileM + rIt + 8 * g;
    int rowc = row < M ? row : M - 1;
    c[rIt] = Cinit ? Cinit[(size_t)rowc * ncols + col] : 0.0f;
  }

  int rowA = tileM + lr;
  int rowAc = rowA < M ? rowA : M - 1;
  for (int kk = 0; kk < kdim; kk += 4) {
    int ka = kk + 2 * g;
    v2f a, b;
    a.x = Amat[(size_t)rowAc * lda + ka];
    a.y = Amat[(size_t)rowAc * lda + ka + 1];
    b.x = Bmat[(size_t)ka * ldb + col];
    b.y = Bmat[(size_t)(ka + 1) * ldb + col];
    // (neg_a, A, neg_b, B, c_mod, C, reuse_a, reuse_b)
    c = __builtin_amdgcn_wmma_f32_16x16x4_f32(false, a, false, b, (short)0, c,
                                              false, false);
  }

#pragma unroll
  for (int rIt = 0; rIt < 8; rIt++) {
    int row = tileM + rIt + 8 * g;
    if (row < M) {
      float v = c[rIt];
      if (apply_silu) v = v / (1.0f + expf(-v));  // silu(x)=x*sigmoid(x)
      Dmat[(size_t)row * ncols + col] = v;
    }
  }
}

// ---------------------------------------------------------------------------
// e_i = dot(h2[i], w_out) + comp[species[i]]; block-reduce; atomicAdd scalar.
// ---------------------------------------------------------------------------
__global__ void energy_reduce(const float* __restrict__ h2,
                              const float* __restrict__ w_out,
                              const float* __restrict__ comp,
                              const int* __restrict__ species,
                              float* __restrict__ out, int n_atoms) {
  __shared__ float red[256];
  int i = blockIdx.x * blockDim.x + threadIdx.x;
  float e = 0.0f;
  if (i < n_atoms) {
    float acc = 0.0f;
#pragma unroll 8
    for (int k = 0; k < 128; k++) acc += h2[(size_t)i * 128 + k] * w_out[k];
    e = acc + comp[species[i]];
  }
  red[threadIdx.x] = e;
  __syncthreads();
  for (int s = 128; s > 0; s >>= 1) {
    if ((int)threadIdx.x < s) red[threadIdx.x] += red[threadIdx.x + s];
    __syncthreads();
  }
  if (threadIdx.x == 0) atomicAdd(out, red[0]);
}

// ---------------------------------------------------------------------------
// Launch. Input order = setup_inputs() dict order:
// 0 positions, 1 emb, 2 species, 3 senders, 4 receivers,
// 5 Wr0, 6 Vr0, 7 Wr1, 8 Vr1, 9 Wr2, 10 Vr2, 11 Wr3, 12 Vr3,
// 13 M1, 14 N1, 15 M2, 16 N2, 17 M3, 18 N3, 19 W1, 20 W2, 21 w_out, 22 comp
// (M1..M3 feed dead code `f` in the reference; skipped.)
// ---------------------------------------------------------------------------
extern "C" void kernel_launch(void* const* d_in, const int* in_sizes, int n_in,
                              void* d_out, int out_size, void* d_ws,
                              size_t ws_size, hipStream_t stream) {
  const float* pos       = (const float*)d_in[0];
  const float* emb       = (const float*)d_in[1];
  const int*   species   = (const int*)d_in[2];
  const int*   senders   = (const int*)d_in[3];
  const int*   receivers = (const int*)d_in[4];
  const float* Wr0 = (const float*)d_in[5];
  const float* Vr0 = (const float*)d_in[6];
  const float* Wr1 = (const float*)d_in[7];
  const float* Vr1 = (const float*)d_in[8];
  const float* Wr2 = (const float*)d_in[9];
  const float* Vr2 = (const float*)d_in[10];
  const float* Wr3 = (const float*)d_in[11];
  const float* Vr3 = (const float*)d_in[12];
  const float* N1  = (const float*)d_in[14];
  const float* N2  = (const float*)d_in[16];
  const float* N3  = (const float*)d_in[18];
  const float* W1  = (const float*)d_in[19];
  const float* W2  = (const float*)d_in[20];
  const float* wo  = (const float*)d_in[21];
  const float* comp = (const float*)d_in[22];

  const int n_atoms = in_sizes[0] / 3;   // 10000
  const int n_edges = in_sizes[3];       // 320000

  float* ws = (float*)d_ws;
  const size_t nA = (size_t)n_atoms;
  float* Abase = ws;                        // n*960 floats
  float* Bbase = ws + nA * 960;             // n*960
  float* invp  = ws + nA * 1920;            // n*192
  float* f2    = ws + nA * 2112;            // n*128
  float* h1    = ws + nA * 2240;            // n*128
  float* h2    = ws + nA * 2368;            // n*128
  float* ncat  = ws + nA * 2496;            // 192*128

  // Zero scatter targets (A+B) and output scalar.
  hipMemsetAsync(Abase, 0, nA * 1920 * sizeof(float), stream);
  hipMemsetAsync(d_out, 0, sizeof(float), stream);

  pack_ncat<<<(192 * 128 + 255) / 256, 256, 0, stream>>>(N1, N2, N3, ncat);

  int eblocks = (n_edges + EDGES_PER_BLOCK - 1) / EDGES_PER_BLOCK;
  edge_pass1<<<eblocks, 256, 0, stream>>>(pos, emb, species, senders, receivers,
                                          Wr0, Wr1, Wr2, Wr3, Abase,
                                          n_atoms, n_edges);
  edge_pass2<<<eblocks, 256, 0, stream>>>(pos, emb, species, senders, receivers,
                                          Vr0, Vr1, Vr2, Vr3, Abase, Bbase,
                                          n_atoms, n_edges);

  inv_kernel<<<(n_atoms * 192 + 255) / 256, 256, 0, stream>>>(Bbase, invp,
                                                              n_atoms);

  int mtiles = (n_atoms + 15) / 16;
  // f2 = B0 + inv @ Ncat   (B0 = Bbase[:, 0, :] laid out [n,128])
  wmma_gemm_f32<<<mtiles, 256, 0, stream>>>(invp, 192, 192, ncat, 128, Bbase,
                                            f2, n_atoms, 128, 0);
  // h1 = silu(f2 @ W1)
  wmma_gemm_f32<<<mtiles, 256, 0, stream>>>(f2, 128, 128, W1, 128, nullptr,
                                            h1, n_atoms, 128, 1);
  // h2 = h1 @ W2
  wmma_gemm_f32<<<mtiles, 256, 0, stream>>>(h1, 128, 128, W2, 128, nullptr,
                                            h2, n_atoms, 128, 0);

  energy_reduce<<<(n_atoms + 255) / 256, 256, 0, stream>>>(
      h2, wo, comp, species, (float*)d_out, n_atoms);
}